// Bspline_segment_calc_14542759264424
// MI455X (gfx1250) — compile-verified
//
#include <hip/hip_runtime.h>
#include <stdint.h>

// ---------------------------------------------------------------------------
// Cubic B-spline basis (k=3) on the reference's fixed extended uniform grid
// (the `grid` input is dead in the reference): knots g_i=(i-3)*0.2-1, i=0..16.
// Output (5, 13, cols) fp32.
//
// Roofline: ~140MB traffic -> ~6us @ 23.3TB/s. VALU was the limiter (~300
// ops/x); this version packs 2 columns per lane (V_PK_*_F32 VOP3P packed fp32)
// and reduces the recurrence to p/(1-p) form -> ~135 ops/x, back under the
// memory bound. CDNA5 paths: TDM tensor_load_to_lds double-buffered staging
// with s_wait_tensorcnt, wave-uniform descriptors from wave 0, NT streaming
// stores (130MB write-once output vs 192MB L2).
// ---------------------------------------------------------------------------

#define ROWS   5
#define DEG    3
#define NOUT   13      // 16 - DEG basis functions survive
#define TILE_E 512     // floats per tile = 2 per thread
#define TPB    2       // tiles per block (double buffered)

#if __has_builtin(__builtin_amdgcn_tensor_load_to_lds) && \
    __has_builtin(__builtin_amdgcn_s_wait_tensorcnt)
#define HAVE_TDM 1
#else
#define HAVE_TDM 0
#endif

typedef uint32_t v4u __attribute__((ext_vector_type(4)));
typedef int      v8i __attribute__((ext_vector_type(8)));
typedef int      v4i __attribute__((ext_vector_type(4)));
typedef float    v2f __attribute__((ext_vector_type(2)));

__device__ __forceinline__ uint32_t rfl_u32(uint32_t v) {
    return (uint32_t)__builtin_amdgcn_readfirstlane((int)v);
}

#if HAVE_TDM
// One TDM load of TILE_E fp32 elements (n_valid real, tail zero-filled via
// tensor_dim0 OOB semantics) from gptr into LDS. Wave-uniform; TDM ignores EXEC.
__device__ __forceinline__ void tdm_load_f32(uint32_t lds_byte_addr,
                                             const float* gptr, int n_valid) {
    uint64_t ga   = (uint64_t)(uintptr_t)gptr;
    uint32_t galo = rfl_u32((uint32_t)(ga & 0xFFFFFFFFull));
    uint32_t gahi = rfl_u32((uint32_t)((ga >> 32) & 0x1FFFFFFull));
    uint32_t nv   = rfl_u32((uint32_t)n_valid);
    uint32_t lab  = rfl_u32(lds_byte_addr);

    v4u g0;                           // D# group 0 (128b)
    g0[0] = 1u;                       // count=1, is_restore=0, gather off
    g0[1] = lab;                      // LDS byte address
    g0[2] = galo;                     // global_addr[31:0]
    g0[3] = gahi | (2u << 30);        // global_addr[56:32] | type=2 ("image")

    v8i g1;                           // D# group 1 (256b)
    g1[0] = (int)(2u << 16);                               // data_size=4B
    g1[1] = (int)((nv & 0xFFFFu) << 16);                   // tensor_dim0[15:0]
    g1[2] = (int)(((nv >> 16) & 0xFFFFu) | (1u << 16));    // td0 hi | tensor_dim1=1
    g1[3] = (int)((uint32_t)TILE_E << 16);                 // tile_dim0 = TILE_E
    g1[4] = 1;                                             // tile_dim1=1, tile_dim2=0
    g1[5] = (int)nv;                                       // tensor_dim0_stride lo
    g1[6] = 0;
    g1[7] = 0;

    v4i g2 = {0, 0, 0, 0};            // <=2D tensor: groups 2/3 unused
    v4i g3 = {0, 0, 0, 0};

#if defined(__clang_major__) && (__clang_major__ >= 23)
    v8i z8 = {0, 0, 0, 0, 0, 0, 0, 0};
    __builtin_amdgcn_tensor_load_to_lds(g0, g1, g2, g3, z8, 0);
#else
    __builtin_amdgcn_tensor_load_to_lds(g0, g1, g2, g3, 0);
#endif
}
#endif // HAVE_TDM

// Knot value, constant-folded for literal i (matches reference to <=1 ulp).
__device__ __forceinline__ float gk(int i) {
    return (float)(i - 3) * 0.2f + (-1.0f);
}

// Packed (2 columns/lane) branch-free Cox–de Boor recurrence.
// Degree-0 indicators use exact knot comparisons; levels use the uniform-knot
// reduction  B[i] = p_i*B[i] + (1-p_{i+1})*B[i+1],  p_i = (x-g_i)*5/j.
__device__ __forceinline__ void bspline_eval2(v2f xv, v2f B[16]) {
    v2f c[17];
#pragma unroll
    for (int i = 0; i < 17; ++i) {
        const float g = gk(i);
        c[i].x = (xv.x >= g) ? 1.0f : 0.0f;
        c[i].y = (xv.y >= g) ? 1.0f : 0.0f;
    }
#pragma unroll
    for (int i = 0; i < 16; ++i)
        B[i] = c[i] - c[i + 1];              // == (x>=g_i)&(x<g_{i+1})

    v2f s[16];                               // s_i = (x - g_i)/h,  h = 0.2f
#pragma unroll
    for (int i = 0; i < 16; ++i)
        s[i] = (xv - gk(i)) * 5.0f;

    // level 1: p = s
#pragma unroll
    for (int i = 0; i < 15; ++i)
        B[i] = s[i] * B[i] + (1.0f - s[i + 1]) * B[i + 1];
    // level 2: p = s/2
#pragma unroll
    for (int i = 0; i < 14; ++i)
        B[i] = (s[i] * 0.5f) * B[i] + (1.0f - s[i + 1] * 0.5f) * B[i + 1];
    // level 3: p = s/3
    const float r3 = 1.0f / 3.0f;
#pragma unroll
    for (int i = 0; i < 13; ++i)
        B[i] = (s[i] * r3) * B[i] + (1.0f - s[i + 1] * r3) * B[i + 1];
}

__global__ __launch_bounds__(256) void bspline_tdm_pk_kernel(
        const float* __restrict__ x, float* __restrict__ out, int cols) {
    __shared__ __align__(16) float tilebuf[2][TILE_E];

    const int tid     = threadIdx.x;
    const int row     = blockIdx.y;
    const int colBase = blockIdx.x * (TILE_E * TPB);
    const float* xrow = x + (size_t)row * (size_t)cols;
    const bool vecStores = ((cols & 1) == 0);   // uniform; cols=500000 -> true

#if HAVE_TDM
    const bool wave0 = (tid < 32);              // wave-uniform (wave32)
    if (wave0) {
        int rem = cols - colBase;
        if (rem > 0)
            tdm_load_f32((uint32_t)(uintptr_t)(&tilebuf[0][0]),
                         xrow + colBase, rem > TILE_E ? TILE_E : rem);
    }
#endif

#pragma unroll
    for (int t = 0; t < TPB; ++t) {
#if HAVE_TDM
        if (wave0) {
            if (t + 1 < TPB) {
                const int off = colBase + (t + 1) * TILE_E;
                const int rem = cols - off;
                if (rem > 0) {
                    tdm_load_f32((uint32_t)(uintptr_t)(&tilebuf[(t + 1) & 1][0]),
                                 xrow + off, rem > TILE_E ? TILE_E : rem);
                    __builtin_amdgcn_s_wait_tensorcnt((short)1);  // current tile done
                } else {
                    __builtin_amdgcn_s_wait_tensorcnt((short)0);
                }
            } else {
                __builtin_amdgcn_s_wait_tensorcnt((short)0);
            }
        }
        __syncthreads();
        const int col2 = colBase + t * TILE_E + 2 * tid;
        const v2f  xv  = *(const v2f*)&tilebuf[t & 1][2 * tid];
#else
        const int col2 = colBase + t * TILE_E + 2 * tid;
        v2f xv;
        xv.x = (col2     < cols) ? xrow[col2]     : 0.0f;
        xv.y = (col2 + 1 < cols) ? xrow[col2 + 1] : 0.0f;
#endif
        v2f B[16];
        bspline_eval2(xv, B);

        if (col2 < cols) {
            const size_t base = (size_t)row * NOUT * (size_t)cols + (size_t)col2;
            if (vecStores && (col2 + 1 < cols)) {
#pragma unroll
                for (int i = 0; i < NOUT; ++i)
                    __builtin_nontemporal_store(B[i],
                        (v2f*)(out + base + (size_t)i * (size_t)cols));
            } else {
#pragma unroll
                for (int i = 0; i < NOUT; ++i) {
                    const size_t o = base + (size_t)i * (size_t)cols;
                    __builtin_nontemporal_store(B[i].x, out + o);
                    if (col2 + 1 < cols)
                        __builtin_nontemporal_store(B[i].y, out + o + 1);
                }
            }
        }
        __syncthreads();
    }
}

extern "C" void kernel_launch(void* const* d_in, const int* in_sizes, int n_in,
                              void* d_out, int out_size, void* d_ws, size_t ws_size,
                              hipStream_t stream) {
    (void)n_in; (void)out_size; (void)d_ws; (void)ws_size;
    const float* x   = (const float*)d_in[0];   // (5, cols) fp32
    // d_in[1] (grid) and d_in[2] (k) are unused by the reference computation.
    float* out = (float*)d_out;                 // (5, 13, cols) fp32
    const int cols = in_sizes[0] / ROWS;

    dim3 block(256);
    dim3 grid((cols + TILE_E * TPB - 1) / (TILE_E * TPB), ROWS);
    bspline_tdm_pk_kernel<<<grid, block, 0, stream>>>(x, out, cols);
}